// SampleLoss_50611894616408
// MI455X (gfx1250) — compile-verified
//
#include <hip/hip_runtime.h>
#include <math.h>
#include <stdint.h>

typedef float v2f __attribute__((ext_vector_type(2)));
typedef float v8f __attribute__((ext_vector_type(8)));

#define N_ROWS 4096
#define DIM    512
#define KPOS   5
#define KNEG   10
#define ROWS5  (N_ROWS * KPOS)   // 20480
#define INV_TEMP 10.0f           // 1 / 0.1

// ---------------------------------------------------------------------------
// Kernel 1: row-wise L2 normalize. blockIdx [0,8192): low 4096 = anchor
// (eps 1e-8), high 4096 = sample (eps 1e-12). 256 threads, 2 elems/thread.
// ---------------------------------------------------------------------------
__global__ void __launch_bounds__(256)
normalize_rows(const float* __restrict__ anchor, const float* __restrict__ sample,
               float* __restrict__ aN, float* __restrict__ sN) {
    const int b    = blockIdx.x;
    const int mat  = b >> 12;
    const int row  = b & (N_ROWS - 1);
    const float* in  = mat ? sample : anchor;
    float*       out = mat ? sN     : aN;
    const float  eps = mat ? 1e-12f : 1e-8f;

    const int tid = threadIdx.x;
    const float x0 = in[row * DIM + tid];
    const float x1 = in[row * DIM + tid + 256];
    float ss = x0 * x0 + x1 * x1;
    #pragma unroll
    for (int off = 16; off > 0; off >>= 1) ss += __shfl_xor(ss, off, 32);

    __shared__ float wss[8];
    const int lane = tid & 31, wave = tid >> 5;
    if (lane == 0) wss[wave] = ss;
    __syncthreads();
    float tot = 0.f;
    #pragma unroll
    for (int w = 0; w < 8; ++w) tot += wss[w];

    const float inv = 1.0f / fmaxf(sqrtf(tot), eps);
    out[row * DIM + tid]       = x0 * inv;
    out[row * DIM + tid + 256] = x1 * inv;
}

// ---------------------------------------------------------------------------
// Kernel 2: sim = aN @ aN^T in 16x16 tiles via V_WMMA_F32_16X16X4_F32.
// One block per 16-row stripe. 8 waves, each computing TWO 16-col tiles per
// chunk (A operand reused across both WMMAs). A stripe staged into LDS via
// async global->LDS b128 copies (ASYNCcnt). All 256 threads run the streaming
// per-row top-5 / bottom-10 selection; sorted candidates merged at the end.
// ---------------------------------------------------------------------------
#define APAD 516             // LDS row stride (floats) for A stripe
#define SPAD 257             // LDS row stride (floats) for sim chunk
#define CHUNK_COLS 256       // 8 waves * 2 tiles * 16 cols
#define NCHUNK (N_ROWS / CHUNK_COLS)  // 16
#define SMEM_BYTES 49472     // max(16*516*4 + 16*257*4, merge arrays 30720)

__global__ void __launch_bounds__(256)
sim_topk(const float* __restrict__ aN, int* __restrict__ posIdx, int* __restrict__ negIdx) {
    __shared__ __align__(16) char smem[SMEM_BYTES];
    float* aLDS   = (float*)smem;            // 16 x 516 floats = 33,024 B
    float* simLDS = (float*)(smem + 33024);  // 16 x 257 floats = 16,448 B

    const int tid  = threadIdx.x;
    const int lane = tid & 31;
    const int wave = tid >> 5;
    const int rowBase = blockIdx.x * 16;

    // ---- Stage the 16x512 A stripe into LDS with async b128 copies. ----
    // 2048 16B vectors total; 8 per thread. Row stride 2064 B (16B aligned).
    #pragma unroll
    for (int it = 0; it < 8; ++it) {
        const int v  = tid + it * 256;        // 0..2047
        const int r  = v >> 7;                // row 0..15
        const int c4 = v & 127;               // 16B-vector within row
        const float* gp = aN + (size_t)(rowBase + r) * DIM + c4 * 4;
        const unsigned ldsOff = (unsigned)(uintptr_t)(aLDS + r * APAD + c4 * 4);
        asm volatile("global_load_async_to_lds_b128 %0, %1, off"
                     :: "v"(ldsOff), "v"(gp) : "memory");
    }
    asm volatile("s_wait_asynccnt 0x0" ::: "memory");
    __syncthreads();

    // WMMA f32 16x16x4 operand layout (wave32):
    //   A: lane&15 = M, (lane>>4)*2 = K-pair base, 2 VGPRs (v2f)
    //   B: lane&15 = N, same K-pair split
    //   C/D: 8 VGPRs, row = v + 8*(lane>>4), col = lane&15
    const int m    = lane & 15;
    const int koff = (lane >> 4) * 2;
    const float* aRow = &aLDS[m * APAD + koff];

    // Per-thread streaming top-k state. Thread (row = tid&15, sub = tid>>4)
    // scans cols [sub*16, sub*16+16) of each chunk for its row.
    const int srow = tid & 15;
    const int sub  = tid >> 4;
    float pv[KPOS]; int pi[KPOS];
    float nv[KNEG]; int ni[KNEG];
    #pragma unroll
    for (int s = 0; s < KPOS; ++s) { pv[s] = -3.0e38f; pi[s] = 0; }
    #pragma unroll
    for (int s = 0; s < KNEG; ++s) { nv[s] =  3.0e38f; ni[s] = 0; }

    for (int chunk = 0; chunk < NCHUNK; ++chunk) {
        const int colBase = chunk * CHUNK_COLS + wave * 32;
        const float* bRow0 = &aN[(size_t)(colBase + m) * DIM + koff];
        const float* bRow1 = &aN[(size_t)(colBase + 16 + m) * DIM + koff];

        v8f acc0 = {}, acc1 = {};
        #pragma unroll 4
        for (int k = 0; k < DIM; k += 4) {
            const v2f a  = *(const v2f*)(aRow + k);
            const v2f b0 = *(const v2f*)(bRow0 + k);
            const v2f b1 = *(const v2f*)(bRow1 + k);
            acc0 = __builtin_amdgcn_wmma_f32_16x16x4_f32(
                false, a, false, b0, (short)0, acc0, false, false);
            acc1 = __builtin_amdgcn_wmma_f32_16x16x4_f32(
                false, a, false, b1, (short)0, acc1, false, false);
        }

        #pragma unroll
        for (int v = 0; v < 8; ++v) {
            const int r = v + 8 * (lane >> 4);
            simLDS[r * SPAD + wave * 32 + m]      = acc0[v];
            simLDS[r * SPAD + wave * 32 + 16 + m] = acc1[v];
        }
        __syncthreads();

        // Streaming selection: ascending index order per thread, strict
        // compares -> ties break toward smaller index (jax.lax.top_k rule).
        {
            const float* sp = &simLDS[srow * SPAD + sub * 16];
            const int gbase = chunk * CHUNK_COLS + sub * 16;
            #pragma unroll
            for (int j = 0; j < 16; ++j) {
                const float v = sp[j];
                const int  gj = gbase + j;
                if (v > pv[KPOS - 1]) {
                    pv[KPOS - 1] = v; pi[KPOS - 1] = gj;
                    #pragma unroll
                    for (int s = KPOS - 1; s > 0; --s) {
                        if (pv[s] > pv[s - 1]) {
                            const float tv = pv[s]; pv[s] = pv[s - 1]; pv[s - 1] = tv;
                            const int   ti = pi[s]; pi[s] = pi[s - 1]; pi[s - 1] = ti;
                        }
                    }
                }
                if (v < nv[KNEG - 1]) {
                    nv[KNEG - 1] = v; ni[KNEG - 1] = gj;
                    #pragma unroll
                    for (int s = KNEG - 1; s > 0; --s) {
                        if (nv[s] < nv[s - 1]) {
                            const float tv = nv[s]; nv[s] = nv[s - 1]; nv[s - 1] = tv;
                            const int   ti = ni[s]; ni[s] = ni[s - 1]; ni[s - 1] = ti;
                        }
                    }
                }
            }
        }
        __syncthreads();
    }

    // ---- Merge phase: candidate arrays alias the (now dead) A/sim LDS. ----
    float* candPV = (float*)smem;            //  5,120 B : [row][sub][5]
    int*   candPI = (int*)(smem + 5120);     //  5,120 B
    float* candNV = (float*)(smem + 10240);  // 10,240 B : [row][sub][10]
    int*   candNI = (int*)(smem + 20480);    // 10,240 B
    {
        const int base = srow * 16 + sub;
        #pragma unroll
        for (int s = 0; s < KPOS; ++s) { candPV[base * KPOS + s] = pv[s]; candPI[base * KPOS + s] = pi[s]; }
        #pragma unroll
        for (int s = 0; s < KNEG; ++s) { candNV[base * KNEG + s] = nv[s]; candNI[base * KNEG + s] = ni[s]; }
    }
    __syncthreads();

    if (tid < 16) {
        const int row = rowBase + tid;
        // 16-way merge of sorted per-sub lists; comparator (v desc, idx asc).
        int headP[16];
        #pragma unroll
        for (int u = 0; u < 16; ++u) headP[u] = 0;
        for (int s = 0; s < KPOS; ++s) {
            float best = -3.4e38f; int bestIdx = 0x7fffffff; int bestSub = 0;
            for (int u = 0; u < 16; ++u) {
                if (headP[u] < KPOS) {
                    const int c = (tid * 16 + u) * KPOS + headP[u];
                    const float v = candPV[c]; const int i = candPI[c];
                    if (v > best || (v == best && i < bestIdx)) { best = v; bestIdx = i; bestSub = u; }
                }
            }
            headP[bestSub]++;
            posIdx[row * KPOS + s] = bestIdx;
        }
        int headN[16];
        #pragma unroll
        for (int u = 0; u < 16; ++u) headN[u] = 0;
        for (int s = 0; s < KNEG; ++s) {
            float best = 3.4e38f; int bestIdx = 0x7fffffff; int bestSub = 0;
            for (int u = 0; u < 16; ++u) {
                if (headN[u] < KNEG) {
                    const int c = (tid * 16 + u) * KNEG + headN[u];
                    const float v = candNV[c]; const int i = candNI[c];
                    if (v < best || (v == best && i < bestIdx)) { best = v; bestIdx = i; bestSub = u; }
                }
            }
            headN[bestSub]++;
            negIdx[row * KNEG + s] = bestIdx;
        }
    }
}

// ---------------------------------------------------------------------------
// Kernel 3: per-logit-row InfoNCE. One wave per r in [0, 5N).
// q = sN[r % N], pos = sN[pos_idx[r/5, r%5]], negs = sN[neg_idx[r/5, :]].
// rowLoss[r] = lse(logits) - logits[0].
// ---------------------------------------------------------------------------
__global__ void __launch_bounds__(256)
infonce_rows(const float* __restrict__ sN, const int* __restrict__ posIdx,
             const int* __restrict__ negIdx, float* __restrict__ rowLoss) {
    const int lane = threadIdx.x & 31;
    const int wave = threadIdx.x >> 5;
    const int r    = blockIdx.x * 8 + wave;          // [0, 20480)
    const int qIdx = r & (N_ROWS - 1);               // r % N
    const int aIdx = r / KPOS;                       // anchor row
    const int slot = r - aIdx * KPOS;                // r % 5

    float q[16];
    const float* qrow = &sN[(size_t)qIdx * DIM];
    #pragma unroll
    for (int t = 0; t < 16; ++t) q[t] = qrow[lane + 32 * t];

    int tgt[1 + KNEG];
    tgt[0] = posIdx[aIdx * KPOS + slot];
    #pragma unroll
    for (int k = 0; k < KNEG; ++k) tgt[1 + k] = negIdx[aIdx * KNEG + k];

    float logits[1 + KNEG];
    #pragma unroll
    for (int c = 0; c < 1 + KNEG; ++c) {
        const float* xrow = &sN[(size_t)tgt[c] * DIM];
        float acc = 0.f;
        #pragma unroll
        for (int t = 0; t < 16; ++t) acc += q[t] * xrow[lane + 32 * t];
        #pragma unroll
        for (int off = 16; off > 0; off >>= 1) acc += __shfl_xor(acc, off, 32);
        logits[c] = acc * INV_TEMP;
    }

    float mx = logits[0];
    #pragma unroll
    for (int c = 1; c < 1 + KNEG; ++c) mx = fmaxf(mx, logits[c]);
    float sum = 0.f;
    #pragma unroll
    for (int c = 0; c < 1 + KNEG; ++c) sum += expf(logits[c] - mx);
    const float lse = mx + logf(sum);

    if (lane == 0) rowLoss[r] = lse - logits[0];
}

// ---------------------------------------------------------------------------
// Kernel 4: deterministic fixed-order reduction of 20480 row losses -> mean.
// ---------------------------------------------------------------------------
__global__ void __launch_bounds__(256)
reduce_loss(const float* __restrict__ rowLoss, float* __restrict__ out) {
    __shared__ float s[256];
    const int tid = threadIdx.x;
    float acc = 0.f;
    for (int i = tid; i < ROWS5; i += 256) acc += rowLoss[i];  // fixed order
    s[tid] = acc;
    __syncthreads();
    for (int stride = 128; stride > 0; stride >>= 1) {
        if (tid < stride) s[tid] += s[tid + stride];
        __syncthreads();
    }
    if (tid == 0) out[0] = s[0] / (float)ROWS5;
}

// ---------------------------------------------------------------------------
extern "C" void kernel_launch(void* const* d_in, const int* in_sizes, int n_in,
                              void* d_out, int out_size, void* d_ws, size_t ws_size,
                              hipStream_t stream) {
    const float* anchor = (const float*)d_in[0];
    const float* sample = (const float*)d_in[1];

    float* aN      = (float*)d_ws;                 //  8 MB
    float* sN      = aN + (size_t)N_ROWS * DIM;    //  8 MB
    int*   posIdx  = (int*)(sN + (size_t)N_ROWS * DIM);
    int*   negIdx  = posIdx + N_ROWS * KPOS;
    float* rowLoss = (float*)(negIdx + N_ROWS * KNEG);
    float* out     = (float*)d_out;

    normalize_rows<<<2 * N_ROWS, 256, 0, stream>>>(anchor, sample, aN, sN);
    sim_topk<<<N_ROWS / 16, 256, 0, stream>>>(aN, posIdx, negIdx);
    infonce_rows<<<ROWS5 / 8, 256, 0, stream>>>(sN, posIdx, negIdx, rowLoss);
    reduce_loss<<<1, 256, 0, stream>>>(rowLoss, out);
}